// Sparsemax_21079699489371
// MI455X (gfx1250) — compile-verified
//
#include <hip/hip_runtime.h>
#include <cstdint>

// Sparsemax over B=16384 rows of L=1024 f32 (equal segments -> dense [B,L]).
// One workgroup per row: TDM async load -> LDS bitonic sort (desc) ->
// WMMA f32 blocked prefix-sum -> tau -> relu(x - max - tau) -> TDM store.

#define B_SEG 16384
#define L_SEG 1024
#define TPB   256

typedef __attribute__((ext_vector_type(2))) float        v2f;
typedef __attribute__((ext_vector_type(8))) float        v8f;
typedef __attribute__((ext_vector_type(4))) unsigned int v4u;
typedef __attribute__((ext_vector_type(8))) int          v8i;
typedef __attribute__((ext_vector_type(4))) int          v4i;

// Build the D# groups for a 1D row transfer of L_SEG f32 between
// global `gaddr` and LDS offset `lds_off`.
__device__ __forceinline__ void make_row_desc(unsigned long long gaddr,
                                              unsigned int lds_off,
                                              v4u& g0, v8i& g1) {
    // group0: count=1 | lds_addr | global_addr[56:0] | type=2 (bits 127:126)
    g0.x = 1u;                                        // count=1, user mode
    g0.y = lds_off;                                   // LDS byte address
    g0.z = (unsigned int)(gaddr & 0xFFFFFFFFull);     // global addr lo
    g0.w = (unsigned int)((gaddr >> 32) & 0x1FFFFFFull) | (2u << 30);

    // group1: data_size=4B, tensor_dim0=tile_dim0=1024, tensor_dim1=1,
    // tile_dim1=1, stride0=1024, no pad/iterate/multicast.
    g1[0] = (int)(2u << 16);                               // data_size = 2 (4B)
    g1[1] = (int)((L_SEG & 0xFFFFu) << 16);                // tensor_dim0[15:0]
    g1[2] = (int)(((L_SEG >> 16) & 0xFFFFu) | (1u << 16)); // td0 hi | td1 lo=1
    g1[3] = (int)((L_SEG & 0xFFFFu) << 16);                // tile_dim0 = 1024
    g1[4] = 1;                                             // tile_dim1=1, tile_dim2=0
    g1[5] = L_SEG;                                         // stride0 lo32
    g1[6] = 0;                                             // stride0 hi | stride1 lo
    g1[7] = 0;                                             // stride1 hi
}

__global__ __launch_bounds__(TPB)
void sparsemax_row_kernel(const float* __restrict__ x, float* __restrict__ out) {
    __shared__ float orig[L_SEG];      // raw row (TDM load destination)
    __shared__ float srt[L_SEG];       // sort buffer (x - rowmax)
    __shared__ float ps[L_SEG];        // WMMA prefix sums; reused as result buf
    __shared__ float chunkoff[64];     // exclusive scan of 16-chunk totals
    __shared__ float redf[TPB];
    __shared__ int   redi[TPB];
    __shared__ float tau_s;

    const int tid = threadIdx.x;
    const int row = blockIdx.x;
    const unsigned long long in_addr =
        (unsigned long long)(uintptr_t)(x + (size_t)row * L_SEG);
    const unsigned long long out_addr =
        (unsigned long long)(uintptr_t)(out + (size_t)row * L_SEG);

    // ---------------- TDM: async DMA row (4KB) global -> LDS ----------------
    if (tid < 32) {  // wave 0 only; TDM ignores EXEC, one descriptor per row
        v4u g0; v8i g1;
        make_row_desc(in_addr, (unsigned int)(uintptr_t)(&orig[0]), g0, g1);
        v4i g2 = {0, 0, 0, 0};
        v4i g3 = {0, 0, 0, 0};
        v8i g4 = {0, 0, 0, 0, 0, 0, 0, 0};
        __builtin_amdgcn_tensor_load_to_lds(g0, g1, g2, g3, g4, 0);
        __builtin_amdgcn_s_wait_tensorcnt(0);
    }
    __syncthreads();

    // ---------------- row max (LDS tree reduce) ----------------
    float m = -__builtin_inff();
#pragma unroll
    for (int u = 0; u < 4; ++u) m = fmaxf(m, orig[tid + u * TPB]);
    redf[tid] = m;
    __syncthreads();
    for (int s = TPB / 2; s > 0; s >>= 1) {
        if (tid < s) redf[tid] = fmaxf(redf[tid], redf[tid + s]);
        __syncthreads();
    }
    const float mx = redf[0];

    // shifted values into sort buffer
    for (int i = tid; i < L_SEG; i += TPB) srt[i] = orig[i] - mx;
    __syncthreads();

    // ---------------- bitonic sort, descending, 1024 elems ----------------
    for (int k = 2; k <= L_SEG; k <<= 1) {
        for (int j = k >> 1; j > 0; j >>= 1) {
#pragma unroll
            for (int u = 0; u < 4; ++u) {
                const int i   = tid + u * TPB;
                const int ixj = i ^ j;
                if (ixj > i) {
                    const float a = srt[i];
                    const float b = srt[ixj];
                    const bool desc = ((i & k) == 0);
                    if (desc ? (a < b) : (a > b)) { srt[i] = b; srt[ixj] = a; }
                }
            }
            __syncthreads();
        }
    }

    // ------- WMMA f32 blocked scan: P = T(lower-tri ones) x X, 16x16x4 -------
    // Waves 0..3 each scan one 256-elem tile = 16 chunks of 16 (columns of X).
    const int lane = tid & 31;
    if (tid < 128) {
        const int base  = (tid >> 5) * 256;
        const int col   = lane & 15;          // N (chunk index in tile)
        const int mrow  = lane & 15;          // M for A-matrix
        const int khalf = (lane >> 4) << 1;   // K offset: lanes 16-31 hold K+2
        v8f acc = {};
#pragma unroll
        for (int j = 0; j < 4; ++j) {         // K slices of 4
            const int k0 = 4 * j + khalf;
            v2f a, b;
            a.x = (k0     <= mrow) ? 1.0f : 0.0f;   // T[m][k0]
            a.y = (k0 + 1 <= mrow) ? 1.0f : 0.0f;   // T[m][k0+1]
            b.x = srt[base + col * 16 + k0];        // X[k0][col]
            b.y = srt[base + col * 16 + k0 + 1];    // X[k0+1][col]
            acc = __builtin_amdgcn_wmma_f32_16x16x4_f32(
                false, a, false, b, (short)0, acc, false, false);
        }
        // D layout: VGPR g holds M=g (lanes 0-15) / M=8+g (lanes 16-31), N=col
        const int mb = (lane >> 4) << 3;
#pragma unroll
        for (int g = 0; g < 8; ++g)
            ps[base + col * 16 + mb + g] = acc[g];
    }
    __syncthreads();

    // exclusive scan of the 64 chunk totals (tiny, serial)
    if (tid == 0) {
        float run = 0.0f;
        for (int c = 0; c < 64; ++c) { chunkoff[c] = run; run += ps[c * 16 + 15]; }
    }
    __syncthreads();

    // ---------------- support size: count((i+1)*sorted[i] > csum[i]) --------
    int cnt = 0;
#pragma unroll
    for (int u = 0; u < 4; ++u) {
        const int i    = tid + u * TPB;
        const float sv = srt[i];
        const float cs = ps[i] + chunkoff[i >> 4] - 1.0f;  // cumsum - 1
        cnt += ((float)(i + 1) * sv > cs) ? 1 : 0;
    }
    redi[tid] = cnt;
    __syncthreads();
    for (int s = TPB / 2; s > 0; s >>= 1) {
        if (tid < s) redi[tid] += redi[tid + s];
        __syncthreads();
    }
    if (tid == 0) {
        int supp = redi[0];
        int idx  = supp - 1;
        if (idx < 0) idx = 0;
        if (idx > L_SEG - 1) idx = L_SEG - 1;
        const float cs = ps[idx] + chunkoff[idx >> 4] - 1.0f;
        tau_s = cs / (float)(supp > 0 ? supp : 1);
    }
    __syncthreads();

    // -------- result row into LDS (reuse ps), contiguous per thread --------
    const float tau = tau_s;
#pragma unroll
    for (int u = 0; u < 4; ++u) {
        const int i = tid * 4 + u;                 // contiguous -> ds_store_b128
        const float v = orig[i] - mx - tau;
        ps[i] = v > 0.0f ? v : 0.0f;
    }
    __syncthreads();

    // ---------------- TDM: async DMA result row LDS -> global ----------------
    if (tid < 32) {
        v4u g0; v8i g1;
        make_row_desc(out_addr, (unsigned int)(uintptr_t)(&ps[0]), g0, g1);
        v4i g2 = {0, 0, 0, 0};
        v4i g3 = {0, 0, 0, 0};
        v8i g4 = {0, 0, 0, 0, 0, 0, 0, 0};
        __builtin_amdgcn_tensor_store_from_lds(g0, g1, g2, g3, g4, 0);
        __builtin_amdgcn_s_wait_tensorcnt(0);
    }
    // other waves may retire; wave 0 holds LDS alive until the store completes
}

extern "C" void kernel_launch(void* const* d_in, const int* in_sizes, int n_in,
                              void* d_out, int out_size, void* d_ws, size_t ws_size,
                              hipStream_t stream) {
    (void)in_sizes; (void)n_in; (void)d_ws; (void)ws_size; (void)out_size;
    const float* x = (const float*)d_in[0];
    // d_in[1] (batch) is implied by the fixed equal-segment layout; unused.
    float* out = (float*)d_out;
    sparsemax_row_kernel<<<B_SEG, TPB, 0, stream>>>(x, out);
}